// InterAgg_12567074308148
// MI455X (gfx1250) — compile-verified
//
#include <hip/hip_runtime.h>

typedef float v2f __attribute__((ext_vector_type(2)));
typedef float v8f __attribute__((ext_vector_type(8)));

#if defined(__AMDGCN__) && __has_builtin(__builtin_amdgcn_wmma_f32_16x16x4_f32)
#define HAS_WMMA_F32X4 1
#else
#define HAS_WMMA_F32X4 0
#endif

constexpr int N_NODES = 8192;
constexpr int FEATN   = 64;
constexpr int K_NBR   = 32;
constexpr int SAMPLE  = 16;
constexpr int NPOS    = 1024;
constexpr int CATF    = 256;   // FEAT + 3*EMB
constexpr int EMB     = 64;

// ---------------------------------------------------------------------------
// Kernel 0: self-feature gather, center/pos scores, softmax probs.
// One wave (32 lanes) per row; lane handles feature f and f+32.
// Rows [0, 8192) = nodes; rows [8192, 9216) = train_pos.
// ---------------------------------------------------------------------------
__global__ void prep_kernel(const int* __restrict__ nodes,
                            const int* __restrict__ train_pos,
                            const float* __restrict__ feat,
                            const float* __restrict__ clfW,    // [64][2]
                            const float* __restrict__ clfb,    // [2]
                            float* __restrict__ cat_ws,        // [8192][256]
                            float* __restrict__ probs_ws,      // [8192][2]
                            float* __restrict__ center_out,    // [8192][2]
                            float* __restrict__ pos_out)       // [1024][2]
{
    const int row  = blockIdx.x * (blockDim.x >> 5) + (threadIdx.x >> 5);
    const int lane = threadIdx.x & 31;
    if (row >= N_NODES + NPOS) return;
    const bool isPos = (row >= N_NODES);
    const int node = isPos ? train_pos[row - N_NODES] : nodes[row];
    const float* f = feat + (long long)node * FEATN;
    const float x0 = f[lane];
    const float x1 = f[lane + 32];
    if (!isPos) {
        cat_ws[(long long)row * CATF + lane]      = x0;
        cat_ws[(long long)row * CATF + lane + 32] = x1;
    }
    float s0 = x0 * clfW[lane * 2 + 0] + x1 * clfW[(lane + 32) * 2 + 0];
    float s1 = x0 * clfW[lane * 2 + 1] + x1 * clfW[(lane + 32) * 2 + 1];
#pragma unroll
    for (int off = 16; off; off >>= 1) {
        s0 += __shfl_xor(s0, off, 32);
        s1 += __shfl_xor(s1, off, 32);
    }
    if (lane == 0) {
        s0 += clfb[0];
        s1 += clfb[1];
        if (isPos) {
            pos_out[(row - N_NODES) * 2 + 0] = s0;
            pos_out[(row - N_NODES) * 2 + 1] = s1;
        } else {
            center_out[row * 2 + 0] = s0;
            center_out[row * 2 + 1] = s1;
            const float m  = fmaxf(s0, s1);
            const float e0 = __expf(s0 - m);
            const float e1 = __expf(s1 - m);
            const float inv = 1.0f / (e0 + e1);
            probs_ws[row * 2 + 0] = e0 * inv;
            probs_ws[row * 2 + 1] = e1 * inv;
        }
    }
}

// ---------------------------------------------------------------------------
// Kernel 1: per (node, relation) neighbor gather + scoring + stable top-16 +
// means. 128 threads per block. LDS tile padded to 65 floats/row to avoid
// bank conflicts on the column-strided scoring reads.
// ---------------------------------------------------------------------------
__global__ void agg_kernel(const int* __restrict__ adj1,
                           const int* __restrict__ adj2,
                           const int* __restrict__ adj3,
                           const float* __restrict__ feat,
                           const float* __restrict__ clfW,
                           const float* __restrict__ clfb,
                           const float* __restrict__ probs_ws,
                           float* __restrict__ top_out,   // [8192][192]
                           float* __restrict__ raw_out)   // [8192][192]
{
    const int i = blockIdx.x;
    const int r = blockIdx.y;
    const int* adj = (r == 0) ? adj1 : (r == 1) ? adj2 : adj3;

    __shared__ float nf[K_NBR][FEATN + 1];
    __shared__ int   nbr[K_NBR];
    __shared__ float dist[K_NBR];
    __shared__ int   sel[K_NBR];

    const int tid = threadIdx.x;   // 128 threads
    if (tid < K_NBR) nbr[tid] = adj[i * K_NBR + tid];
    __syncthreads();

    // Gather 32 x 64 floats, coalesced per neighbor row.
    for (int idx = tid; idx < K_NBR * FEATN; idx += blockDim.x) {
        const int k = idx >> 6;
        const int f = idx & 63;
        nf[k][f] = feat[(long long)nbr[k] * FEATN + f];
    }
    __syncthreads();

    // Neighbor scoring: softmax over 2 classes, L1 distance to center probs.
    if (tid < K_NBR) {
        float s0 = 0.f, s1 = 0.f;
        for (int f = 0; f < FEATN; ++f) {
            const float v = nf[tid][f];
            s0 += v * clfW[f * 2 + 0];
            s1 += v * clfW[f * 2 + 1];
        }
        s0 += clfb[0];
        s1 += clfb[1];
        const float m  = fmaxf(s0, s1);
        const float e0 = __expf(s0 - m);
        const float e1 = __expf(s1 - m);
        const float inv = 1.0f / (e0 + e1);
        const float p0 = e0 * inv, p1 = e1 * inv;
        const float c0 = probs_ws[i * 2 + 0];
        const float c1 = probs_ws[i * 2 + 1];
        dist[tid] = fabsf(p0 - c0) + fabsf(p1 - c1);
    }
    __syncthreads();

    // Stable top-16 smallest distance (jax.lax.top_k tie-break: lower index
    // wins): rank by lexicographic (dist, index), keep rank < SAMPLE.
    if (tid < K_NBR) {
        const float dk = dist[tid];
        int cnt = 0;
        for (int j = 0; j < K_NBR; ++j) {
            const float dj = dist[j];
            cnt += (dj < dk) || (dj == dk && j < tid);
        }
        sel[tid] = (cnt < SAMPLE) ? 1 : 0;
    }
    __syncthreads();

    // Means over selected / all neighbors.
    if (tid < FEATN) {
        float ra = 0.f, ag = 0.f;
        for (int k = 0; k < K_NBR; ++k) {
            const float v = nf[k][tid];
            ra += v;
            if (sel[k]) ag += v;
        }
        const long long o = (long long)i * 192 + r * 64 + tid;
        top_out[o] = ag * (1.0f / SAMPLE);
        raw_out[o] = ra * (1.0f / K_NBR);
    }
}

// ---------------------------------------------------------------------------
// WMMA helpers: one wave computes one 16x16 f32 output tile, K accumulated
// 4 at a time with V_WMMA_F32_16X16X4_F32.
// A fragment layout (16x4 f32): lane l (0-15) -> {A[l,k0], A[l,k0+1]},
// lane l+16 -> {A[l,k0+2], A[l,k0+3]}  == one aligned float2 per lane.
// B fragment (4x16): vgpr0 = B[k0+2*half][colBase+l], vgpr1 = next K row.
// C/D layout: element v of acc -> (M = v + 8*half, N = colBase + l).
// ---------------------------------------------------------------------------

// Kernel 2a: r_t = relu(agg_r @ W_r), written into cat buffer cols 64+64r.
__global__ void rgemm_kernel(const float* __restrict__ top_feats, // [8192][192]
                             const float* __restrict__ W1,
                             const float* __restrict__ W2,
                             const float* __restrict__ W3,
                             float* __restrict__ cat_ws)          // [8192][256]
{
    const int rowTile = blockIdx.x * 16;
    const int r = blockIdx.y;
    const float* B = (r == 0) ? W1 : (r == 1) ? W2 : W3;   // [64][64]
    const int wave = threadIdx.x >> 5;                     // 4 waves -> 4 col tiles
    const int lane = threadIdx.x & 31;
    const int half = lane >> 4;
    const int l    = lane & 15;
    const int colBase = wave * 16;
    const float* A = top_feats + (long long)(rowTile + l) * 192 + r * 64;

#if HAS_WMMA_F32X4
    v8f acc = {};
    for (int k0 = 0; k0 < EMB; k0 += 4) {
        const int kb = k0 + 2 * half;
        const v2f a = *(const v2f*)(A + kb);
        v2f b;
        b.x = B[kb * 64 + colBase + l];
        b.y = B[(kb + 1) * 64 + colBase + l];
        acc = __builtin_amdgcn_wmma_f32_16x16x4_f32(false, a, false, b,
                                                    (short)0, acc, false, false);
    }
#pragma unroll
    for (int v = 0; v < 8; ++v) {
        const int row = rowTile + v + 8 * half;
        cat_ws[(long long)row * CATF + 64 + r * 64 + colBase + l] =
            fmaxf(acc[v], 0.0f);
    }
#else
    float acc[8] = {};
    for (int k = 0; k < EMB; ++k) {
        const float bk = B[k * 64 + colBase + l];
        for (int v = 0; v < 8; ++v)
            acc[v] += top_feats[(long long)(rowTile + v + 8 * half) * 192 + r * 64 + k] * bk;
    }
    for (int v = 0; v < 8; ++v) {
        const int row = rowTile + v + 8 * half;
        cat_ws[(long long)row * CATF + 64 + r * 64 + colBase + l] =
            fmaxf(acc[v], 0.0f);
    }
#endif
}

// Kernel 2b: combined = relu(cat @ weight).T  -> [64][8192]
__global__ void cgemm_kernel(const float* __restrict__ cat_ws,  // [8192][256]
                             const float* __restrict__ weight,  // [256][64]
                             float* __restrict__ comb_out)      // [64][8192]
{
    const int rowTile = blockIdx.x * 16;
    const int wave = threadIdx.x >> 5;
    const int lane = threadIdx.x & 31;
    const int half = lane >> 4;
    const int l    = lane & 15;
    const int colBase = wave * 16;
    const float* A = cat_ws + (long long)(rowTile + l) * CATF;

#if HAS_WMMA_F32X4
    v8f acc = {};
    for (int k0 = 0; k0 < CATF; k0 += 4) {
        const int kb = k0 + 2 * half;
        const v2f a = *(const v2f*)(A + kb);
        v2f b;
        b.x = weight[kb * 64 + colBase + l];
        b.y = weight[(kb + 1) * 64 + colBase + l];
        acc = __builtin_amdgcn_wmma_f32_16x16x4_f32(false, a, false, b,
                                                    (short)0, acc, false, false);
    }
#pragma unroll
    for (int v = 0; v < 8; ++v) {
        const int row = rowTile + v + 8 * half;
        const int col = colBase + l;
        comb_out[(long long)col * N_NODES + row] = fmaxf(acc[v], 0.0f);
    }
#else
    float acc[8] = {};
    for (int k = 0; k < CATF; ++k) {
        const float bk = weight[k * 64 + colBase + l];
        for (int v = 0; v < 8; ++v)
            acc[v] += cat_ws[(long long)(rowTile + v + 8 * half) * CATF + k] * bk;
    }
    for (int v = 0; v < 8; ++v) {
        const int row = rowTile + v + 8 * half;
        const int col = colBase + l;
        comb_out[(long long)col * N_NODES + row] = fmaxf(acc[v], 0.0f);
    }
#endif
}

// ---------------------------------------------------------------------------
extern "C" void kernel_launch(void* const* d_in, const int* in_sizes, int n_in,
                              void* d_out, int out_size, void* d_ws, size_t ws_size,
                              hipStream_t stream) {
    (void)in_sizes; (void)n_in; (void)out_size; (void)ws_size;

    const int*   nodes     = (const int*)d_in[0];
    // d_in[1] labels: unused by the reference outputs
    const int*   adj1      = (const int*)d_in[2];
    const int*   adj2      = (const int*)d_in[3];
    const int*   adj3      = (const int*)d_in[4];
    const int*   train_pos = (const int*)d_in[5];
    const float* feat      = (const float*)d_in[6];
    // d_in[7] pe_table: dead code (sel_pe never returned)
    const float* clfW      = (const float*)d_in[8];
    const float* clfb      = (const float*)d_in[9];
    const float* W1        = (const float*)d_in[10];
    const float* W2        = (const float*)d_in[11];
    const float* W3        = (const float*)d_in[12];
    const float* weight    = (const float*)d_in[13];

    float* out    = (float*)d_out;
    float* comb   = out;                       // [64][8192]   524288
    float* center = out + 524288;              // [8192][2]     16384
    float* topf   = out + 540672;              // [8192][192] 1572864
    float* rawf   = out + 2113536;             // [8192][192] 1572864
    float* posf   = out + 3686400;             // [1024][2]      2048

    float* cat_ws   = (float*)d_ws;                          // [8192][256]
    float* probs_ws = cat_ws + (size_t)N_NODES * CATF;       // [8192][2]

    prep_kernel<<<dim3((N_NODES + NPOS) / 8), 256, 0, stream>>>(
        nodes, train_pos, feat, clfW, clfb, cat_ws, probs_ws, center, posf);

    agg_kernel<<<dim3(N_NODES, 3), 128, 0, stream>>>(
        adj1, adj2, adj3, feat, clfW, clfb, probs_ws, topf, rawf);

    rgemm_kernel<<<dim3(N_NODES / 16, 3), 128, 0, stream>>>(
        topf, W1, W2, W3, cat_ws);

    cgemm_kernel<<<dim3(N_NODES / 16), 128, 0, stream>>>(
        cat_ws, weight, comb);
}